// GATModel_59081570124183
// MI455X (gfx1250) — compile-verified
//
#include <hip/hip_runtime.h>
#include <hip/hip_bf16.h>

// ---------------------------------------------------------------------------
// GAT forward for MI455X (gfx1250, wave32, WMMA + TDM).
//   Phase 1 (WMMA): emb = X @ W  fused with per-node/per-head attention scores
//                   W tile staged into LDS via the Tensor Data Mover.
//   Phase 2: edge softmax numerators + segment-sum denominator (f32 atomics)
//   Phase 3: attention-weighted scatter aggregate into out (f32 atomics)
// All intermediates (~112 MB) fit in the 192 MB L2, so the irregular edge
// phases run at L2 bandwidth rather than HBM.
// ---------------------------------------------------------------------------

typedef __attribute__((ext_vector_type(2))) float        v2f;
typedef __attribute__((ext_vector_type(8))) float        v8f;
typedef __attribute__((ext_vector_type(4))) unsigned int v4ui;
typedef __attribute__((ext_vector_type(4))) int          v4si;
typedef __attribute__((ext_vector_type(8))) int          v8si;

#define IN_C   128
#define OUT_C  128
#define HEADS  8
#define HEAD_C 16
#define NEG_SLOPE 0.2f
#define GAT_EPS 1e-16f

#ifndef __has_builtin
#define __has_builtin(x) 0
#endif
#if __has_builtin(__builtin_amdgcn_tensor_load_to_lds) && \
    __has_builtin(__builtin_amdgcn_s_wait_tensorcnt)
#define GAT_USE_TDM 1
#else
#define GAT_USE_TDM 0
#endif

// ---------------------------------------------------------------------------
// Init: out[n, c] = bias[c]  (folds the final bias add; atomics accumulate on
// top), and zero the softmax denominator. Re-runs every launch (graph replay
// safe, no stale state).
// ---------------------------------------------------------------------------
__global__ void gat_init_kernel(float* __restrict__ out,
                                const float* __restrict__ bias,
                                float* __restrict__ denom,
                                long long total, long long denom_n) {
    long long idx = (long long)blockIdx.x * blockDim.x + threadIdx.x;
    if (idx < total)   out[idx] = bias[idx & (OUT_C - 1)];
    if (idx < denom_n) denom[idx] = 0.0f;
}

// ---------------------------------------------------------------------------
// Fused GEMM + scores. One wave32 per 16-node row tile; 8 waves per block.
// W (128x128 fp32 = 64 KB) staged in LDS once per workgroup — by TDM when the
// builtin is available (one DMA op + s_wait_tensorcnt), else by float4 copies.
//
// 32-bit WMMA A layout (16x4):  lanes 0-15 hold M=0..15 {K0,K1},
//                               lanes 16-31 hold M=0..15 {K2,K3}.
// 32-bit WMMA B layout (4x16):  lanes 0-15 hold N=0..15 {K0,K1},
//                               lanes 16-31 hold N=0..15 {K2,K3}.
// 32-bit C/D layout (16x16):    VGPR r: lanes 0-15 -> (M=r,   N=lane),
//                                       lanes 16-31 -> (M=r+8, N=lane-16).
// ---------------------------------------------------------------------------
__global__ void __launch_bounds__(256)
gat_gemm_scores_kernel(const float* __restrict__ X,
                       const float* __restrict__ W,
                       const float* __restrict__ a_left,
                       const float* __restrict__ a_right,
                       float* __restrict__ emb,
                       float* __restrict__ src_scores,
                       float* __restrict__ tgt_scores,
                       int N) {
    __shared__ float Wl[IN_C * OUT_C];   // 64 KB

    const int tid = threadIdx.x;

#if GAT_USE_TDM
    // --- Tensor Data Mover: DMA the whole 128x128 fp32 W tile into LDS. ---
    if (tid < 32) {   // wave 0 issues; TDM ignores EXEC, one op per wave
        const unsigned long long gaddr = (unsigned long long)(const void*)W;
        // Low 32 bits of the flat shared-aperture address == LDS byte offset
        // (ISA 10.2).
        const unsigned int lds_off = (unsigned int)(unsigned long long)(void*)&Wl[0];

        v4ui g0;
        g0[0] = 1u;                                   // count=1, user D#
        g0[1] = lds_off;                              // lds_addr (bytes)
        g0[2] = (unsigned int)(gaddr & 0xFFFFFFFFull);        // global_addr lo
        g0[3] = (unsigned int)((gaddr >> 32) & 0x01FFFFFFull) // global_addr hi
              | (2u << 30);                           // type = 2 ("image")

        v8si g1;
        g1[0] = (int)(2u << 16);        // data_size=2 (4 bytes); no multicast
        g1[1] = (int)(128u << 16);      // tensor_dim0[15:0] = 128 (k cols)
        g1[2] = (int)(128u << 16);      // tensor_dim1[15:0] = 128 (rows)
        g1[3] = (int)(128u << 16);      // tile_dim0 = 128
        g1[4] = 128;                    // tile_dim1 = 128
        g1[5] = 128;                    // tensor_dim0_stride[31:0] = 128 elems
        g1[6] = (int)(16384u << 16);    // tensor_dim1_stride[15:0] = 128*128
        g1[7] = 0;                      // tensor_dim1_stride[47:16] = 0

        v4si gz  = {0, 0, 0, 0};        // groups 2/3 unused (2D tensor)
#if __clang_major__ >= 23
        v8si gz8 = {0, 0, 0, 0, 0, 0, 0, 0};
        __builtin_amdgcn_tensor_load_to_lds(g0, g1, gz, gz, gz8, 0);
#else
        __builtin_amdgcn_tensor_load_to_lds(g0, g1, gz, gz, 0);
#endif
        __builtin_amdgcn_s_wait_tensorcnt(0);
    }
    __syncthreads();
#else
    // --- Fallback: float4-vectorized cooperative copy. ---
    {
        const float4* Wv  = (const float4*)W;
        float4*       Wlv = (float4*)Wl;
#pragma unroll
        for (int i = 0; i < 16; ++i)
            Wlv[tid + i * 256] = Wv[tid + i * 256];
    }
    __syncthreads();
#endif

    const int wave = tid >> 5;
    const int lane = tid & 31;
    const int tile = blockIdx.x * 8 + wave;
    const int base = tile * 16;
    if (base >= N) return;                 // uniform per wave: EXEC stays full

    const int half = lane >> 4;            // 0: lanes 0-15, 1: lanes 16-31
    const int col  = lane & 15;            // M for A-frag, N for B-frag
    const int klo  = half << 1;            // K offset within the 4-wide step
    int arow_idx = base + col;
    if (arow_idx >= N) arow_idx = N - 1;   // clamp (N % 16 == 0 here anyway)
    const float* __restrict__ arow = X + (long long)arow_idx * IN_C;

    v8f acc[HEADS];
#pragma unroll
    for (int nt = 0; nt < HEADS; ++nt)
        acc[nt] = (v8f){0.f, 0.f, 0.f, 0.f, 0.f, 0.f, 0.f, 0.f};

    for (int kk = 0; kk < IN_C; kk += 4) {
        if (kk + 32 < IN_C)
            __builtin_prefetch(arow + kk + 32, 0, 3);   // near-scope prefetch

        v2f a;
        a.x = arow[kk + klo];
        a.y = arow[kk + klo + 1];

#pragma unroll
        for (int nt = 0; nt < HEADS; ++nt) {
            v2f b;
            b.x = Wl[(kk + klo)     * OUT_C + nt * HEAD_C + col];
            b.y = Wl[(kk + klo + 1) * OUT_C + nt * HEAD_C + col];
            acc[nt] = __builtin_amdgcn_wmma_f32_16x16x4_f32(
                false, a, false, b, (short)0, acc[nt], false, false);
        }
    }

    // Write emb tile + fused per-head score reduction.
    // Head nt owns output channels [nt*16, nt*16+16); channel-in-head == col.
#pragma unroll
    for (int nt = 0; nt < HEADS; ++nt) {
        const float aLv = a_left [col * HEADS + nt];
        const float aRv = a_right[col * HEADS + nt];
#pragma unroll
        for (int r = 0; r < 8; ++r) {
            const int node = base + r + (half << 3);
            const float v  = acc[nt][r];
            emb[(long long)node * OUT_C + nt * HEAD_C + col] = v;

            float pl = v * aLv;
            float pr = v * aRv;
#pragma unroll
            for (int m = 1; m < 16; m <<= 1) {   // reduce within 16-lane half
                pl += __shfl_xor(pl, m, 32);
                pr += __shfl_xor(pr, m, 32);
            }
            if (col == 0) {
                src_scores[node * HEADS + nt] = pl;
                tgt_scores[node * HEADS + nt] = pr;
            }
        }
    }
}

// ---------------------------------------------------------------------------
// Edge pass 1: exp(leaky_relu(score)) per (edge, head) + segment-sum denom.
// ---------------------------------------------------------------------------
__global__ void gat_edge_softmax_kernel(const int* __restrict__ edge_index,
                                        const float* __restrict__ src_scores,
                                        const float* __restrict__ tgt_scores,
                                        float* __restrict__ exp_scores,
                                        float* __restrict__ denom,
                                        long long E) {
    long long idx = (long long)blockIdx.x * blockDim.x + threadIdx.x;
    if (idx >= E * HEADS) return;
    const long long e = idx >> 3;
    const int h = (int)(idx & 7);
    const int s = edge_index[e];
    const int t = edge_index[E + e];
    float x = src_scores[(long long)s * HEADS + h] +
              tgt_scores[(long long)t * HEADS + h];
    x = (x > 0.0f) ? x : NEG_SLOPE * x;      // leaky_relu
    const float ex = __expf(x);
    exp_scores[idx] = ex;
    atomicAdd(&denom[(long long)t * HEADS + h], ex);
}

// ---------------------------------------------------------------------------
// Edge pass 2: out[t, c] += emb[s, c] * attn[e, c/16].
// One wave32 per edge; each lane owns 4 channels -> float4 (b128) emb gather,
// 4 f32 atomics per lane. Channels c..c+3 never straddle a head boundary.
// ---------------------------------------------------------------------------
__global__ void gat_edge_aggregate_kernel(const int* __restrict__ edge_index,
                                          const float* __restrict__ emb,
                                          const float* __restrict__ exp_scores,
                                          const float* __restrict__ denom,
                                          float* __restrict__ out,
                                          long long E) {
    long long idx = (long long)blockIdx.x * blockDim.x + threadIdx.x;
    const long long e = idx >> 5;
    if (e >= E) return;
    const int lane = (int)(idx & 31);
    const int c    = lane << 2;              // 4 channels per lane
    const int h    = c >> 4;
    const int s = edge_index[e];
    const int t = edge_index[E + e];
    const float attn = exp_scores[e * HEADS + h] /
                       (denom[(long long)t * HEADS + h] + GAT_EPS);
    const float4 v = ((const float4*)emb)[(long long)s * (OUT_C / 4) + lane];
    float* o = out + (long long)t * OUT_C + c;
    atomicAdd(o + 0, v.x * attn);
    atomicAdd(o + 1, v.y * attn);
    atomicAdd(o + 2, v.z * attn);
    atomicAdd(o + 3, v.w * attn);
}

// ---------------------------------------------------------------------------
// Launch
// ---------------------------------------------------------------------------
extern "C" void kernel_launch(void* const* d_in, const int* in_sizes, int n_in,
                              void* d_out, int out_size, void* d_ws, size_t ws_size,
                              hipStream_t stream) {
    const float* X    = (const float*)d_in[0];   // [N, 128]
    const int*   EI   = (const int*)  d_in[1];   // [2, E]
    const float* W    = (const float*)d_in[2];   // [128, 128]
    const float* aL   = (const float*)d_in[3];   // [16, 8]
    const float* aR   = (const float*)d_in[4];   // [16, 8]
    const float* bias = (const float*)d_in[5];   // [128]

    const long long N = in_sizes[0] / IN_C;
    const long long E = in_sizes[1] / 2;

    // Workspace partition
    float* emb        = (float*)d_ws;                    // N*128
    float* src_scores = emb        + N * OUT_C;          // N*8
    float* tgt_scores = src_scores + N * HEADS;          // N*8
    float* denom      = tgt_scores + N * HEADS;          // N*8
    float* exp_scores = denom      + N * HEADS;          // E*8

    float* out = (float*)d_out;

    // 1) init out with bias, zero denom
    {
        const long long total = N * OUT_C;
        const int blocks = (int)((total + 255) / 256);
        gat_init_kernel<<<blocks, 256, 0, stream>>>(out, bias, denom,
                                                    total, N * HEADS);
    }
    // 2) WMMA GEMM + fused scores (W staged via TDM)
    {
        const long long tiles = (N + 15) / 16;
        const int blocks = (int)((tiles + 7) / 8);   // 8 waves per block
        gat_gemm_scores_kernel<<<blocks, 256, 0, stream>>>(
            X, W, aL, aR, emb, src_scores, tgt_scores, (int)N);
    }
    // 3) edge softmax numerators + denominator
    {
        const long long total = E * HEADS;
        const int blocks = (int)((total + 255) / 256);
        gat_edge_softmax_kernel<<<blocks, 256, 0, stream>>>(
            EI, src_scores, tgt_scores, exp_scores, denom, E);
    }
    // 4) weighted scatter aggregate (wave per edge)
    {
        const long long total = E * 32;
        const int blocks = (int)((total + 255) / 256);
        gat_edge_aggregate_kernel<<<blocks, 256, 0, stream>>>(
            EI, emb, exp_scores, denom, out, E);
    }
}